// GraphEvolution_76768245448918
// MI455X (gfx1250) — compile-verified
//
#include <hip/hip_runtime.h>
#include <math.h>

// ---------------- problem constants ----------------
static constexpr int T_   = 4;
static constexpr int N_   = 4096;
static constexpr int M_   = T_ * N_;      // 16384 tokens
static constexpr int HID  = 64;
static constexpr int HC   = 512;          // 8 heads * 64
static constexpr int FF   = 32;
static constexpr int E_   = 131072;
static constexpr int E2_  = E_ + M_;      // edges + self loops

// activation codes for fused GEMM epilogue
#define ACT_NONE 0
#define ACT_RELU 1
#define ACT_GELU 2

typedef float v2f __attribute__((ext_vector_type(2)));
typedef float v8f __attribute__((ext_vector_type(8)));

__device__ __forceinline__ float gelu_f(float x) {
  // exact gelu: x * 0.5 * (1 + erf(x / sqrt(2)))
  return 0.5f * x * (1.0f + erff(x * 0.70710678118654752f));
}

__device__ __forceinline__ float act_apply(float v, int act) {
  if (act == ACT_RELU)      return v > 0.f ? v : 0.f;
  if (act == ACT_GELU)      return gelu_f(v);
  return v;
}

// order-preserving float<->uint map for atomicMax on signed floats
__device__ __forceinline__ unsigned fenc(float f) {
  unsigned u = __float_as_uint(f);
  return (u & 0x80000000u) ? ~u : (u | 0x80000000u);
}
__device__ __forceinline__ float fdec(unsigned u) {
  u = (u & 0x80000000u) ? (u & 0x7fffffffu) : ~u;
  return __uint_as_float(u);
}

// ================= WMMA fp32 GEMM kernels: Y = act(X[M,K] @ W[N,K]^T + b) =================
// Fragment layouts (ISA 7.12.2, V_WMMA_F32_16X16X4_F32):
//  A (16x4, 2 VGPR): m = lane&15, regs hold K = 2*(lane>>4), 2*(lane>>4)+1
//  B (4x16, 2 VGPR): mirror of A with n = lane&15
//  C/D (16x16, 8 VGPR): m = r + 8*(lane>>4), n = lane&15

// ---- blocked kernel: one wave computes a 64x64 tile (4x4 WMMA tiles, 16 f32 accs) ----
// requires Mrows%64==0 && N%64==0 && K%4==0.  16 WMMAs per K-step, ~16 flops/byte reuse.
__global__ void __launch_bounds__(256)
gemm_wmma_f32_blk(const float* __restrict__ X, const float* __restrict__ W,
                  const float* __restrict__ bias, float* __restrict__ Y,
                  int Mrows, int N, int K, int act)
{
  const int wid  = blockIdx.x * (blockDim.x >> 5) + (threadIdx.x >> 5);
  const int lane = threadIdx.x & 31;
  const int ntiles = N >> 6;
  const int mtiles = Mrows >> 6;
  if (wid >= ntiles * mtiles) return;           // uniform per-wave exit, EXEC stays all-1s
  const int tm = wid / ntiles;
  const int tn = wid - tm * ntiles;
  const int m0 = tm << 6;
  const int n0 = tn << 6;
  const int half = lane >> 4;
  const int l15  = lane & 15;

  const float* __restrict__ xr0 = X + (size_t)(m0 + l15) * K + 2 * half;
  const float* __restrict__ wr0 = W + (size_t)(n0 + l15) * K + 2 * half;
  const size_t xstep = (size_t)16 * K;          // next 16-row M sub-tile
  v8f acc[16];
  #pragma unroll
  for (int i = 0; i < 16; ++i) acc[i] = (v8f){0.f,0.f,0.f,0.f,0.f,0.f,0.f,0.f};

  for (int k = 0; k < K; k += 4) {
    // prefetch next K chunk into the near caches (global_prefetch_b8, WGP scope)
    __builtin_prefetch(xr0 + k + 32, 0, 3);
    __builtin_prefetch(wr0 + k + 32, 0, 3);
    v2f a[4], b[4];
    #pragma unroll
    for (int mi = 0; mi < 4; ++mi) {
      a[mi].x = xr0[(size_t)mi * xstep + k];
      a[mi].y = xr0[(size_t)mi * xstep + k + 1];
    }
    #pragma unroll
    for (int ni = 0; ni < 4; ++ni) {
      b[ni].x = wr0[(size_t)ni * xstep + k];
      b[ni].y = wr0[(size_t)ni * xstep + k + 1];
    }
    #pragma unroll
    for (int mi = 0; mi < 4; ++mi)
      #pragma unroll
      for (int ni = 0; ni < 4; ++ni)
        acc[mi * 4 + ni] = __builtin_amdgcn_wmma_f32_16x16x4_f32(
            false, a[mi], false, b[ni], (short)0, acc[mi * 4 + ni], false, false);
  }

  #pragma unroll
  for (int ni = 0; ni < 4; ++ni) {
    const int n = n0 + ni * 16 + l15;
    const float bn = bias ? bias[n] : 0.f;
    #pragma unroll
    for (int mi = 0; mi < 4; ++mi) {
      #pragma unroll
      for (int r = 0; r < 8; ++r) {
        const int m = m0 + mi * 16 + r + 8 * half;
        Y[(size_t)m * N + n] = act_apply(acc[mi * 4 + ni][r] + bn, act);
      }
    }
  }
}

// ---- simple kernel: one wave computes one 16x16 tile (used when N%64 != 0) ----
__global__ void __launch_bounds__(256)
gemm_wmma_f32(const float* __restrict__ X, const float* __restrict__ W,
              const float* __restrict__ bias, float* __restrict__ Y,
              int Mrows, int N, int K, int act)
{
  const int wid  = blockIdx.x * (blockDim.x >> 5) + (threadIdx.x >> 5);
  const int lane = threadIdx.x & 31;
  const int ntiles = N >> 4;
  const int mtiles = Mrows >> 4;
  if (wid >= ntiles * mtiles) return;
  const int tm = wid / ntiles;
  const int tn = wid - tm * ntiles;
  const int m0 = tm << 4;
  const int n0 = tn << 4;
  const int half = lane >> 4;
  const int l15  = lane & 15;

  const float* __restrict__ xrow = X + (size_t)(m0 + l15) * K + 2 * half;
  const float* __restrict__ wrow = W + (size_t)(n0 + l15) * K + 2 * half;

  v8f acc = {0.f,0.f,0.f,0.f,0.f,0.f,0.f,0.f};
  #pragma unroll 4
  for (int k = 0; k < K; k += 4) {
    v2f a, b;
    a.x = xrow[k]; a.y = xrow[k + 1];
    b.x = wrow[k]; b.y = wrow[k + 1];
    acc = __builtin_amdgcn_wmma_f32_16x16x4_f32(false, a, false, b,
                                                (short)0, acc, false, false);
  }

  const int n = n0 + l15;
  const float bn = bias ? bias[n] : 0.f;
  #pragma unroll
  for (int r = 0; r < 8; ++r) {
    const int m = m0 + r + 8 * half;
    Y[(size_t)m * N + n] = act_apply(acc[r] + bn, act);
  }
}

// ---------------- LayerNorm(a + b) over 64 channels, one wave per row ----------------
__global__ void __launch_bounds__(256)
add_ln_kernel(const float* __restrict__ a, const float* __restrict__ b,
              const float* __restrict__ g, const float* __restrict__ be,
              float* __restrict__ out, int Mrows)
{
  const int row  = (blockIdx.x * blockDim.x + threadIdx.x) >> 5;
  const int lane = threadIdx.x & 31;
  if (row >= Mrows) return;
  const float* pa = a + (size_t)row * HID;
  const float* pb = b + (size_t)row * HID;
  float v0 = pa[lane]      + pb[lane];
  float v1 = pa[lane + 32] + pb[lane + 32];
  float s = v0 + v1;
  #pragma unroll
  for (int off = 16; off > 0; off >>= 1) s += __shfl_xor(s, off, 32);
  const float mean = s * (1.f / 64.f);
  const float d0 = v0 - mean, d1 = v1 - mean;
  float vs = d0 * d0 + d1 * d1;
  #pragma unroll
  for (int off = 16; off > 0; off >>= 1) vs += __shfl_xor(vs, off, 32);
  const float inv = rsqrtf(vs * (1.f / 64.f) + 1e-5f);
  float* po = out + (size_t)row * HID;
  po[lane]      = d0 * inv * g[lane]      + be[lane];
  po[lane + 32] = d1 * inv * g[lane + 32] + be[lane + 32];
}

// ---------------- small helpers ----------------
__global__ void build_xin(const float* __restrict__ x, const float* __restrict__ params,
                          float* __restrict__ xin, int Mrows)
{
  int i = blockIdx.x * blockDim.x + threadIdx.x;
  if (i >= Mrows * 8) return;
  int m = i >> 3, c = i & 7;
  xin[i] = (c < 2) ? x[m * 2 + c] : params[c - 2];
}

__global__ void build_edges(const int* __restrict__ ei, int* __restrict__ src,
                            int* __restrict__ dst, int E, int E2)
{
  int i = blockIdx.x * blockDim.x + threadIdx.x;
  if (i >= E2) return;
  if (i < E) { src[i] = ei[i]; dst[i] = ei[E + i]; }
  else       { src[i] = i - E; dst[i] = i - E; }
}

__global__ void reduce_sum(const float* __restrict__ x, float* __restrict__ out, int n)
{
  int i = blockIdx.x * blockDim.x + threadIdx.x;
  float v = (i < n) ? x[i] : 0.f;
  #pragma unroll
  for (int off = 16; off > 0; off >>= 1) v += __shfl_xor(v, off, 32);
  if ((threadIdx.x & 31) == 0) atomicAdd(out, v);
}

__global__ void build_ea(const float* __restrict__ ea_in, const float* __restrict__ rsum,
                         float* __restrict__ ea, int E, int E2)
{
  int i = blockIdx.x * blockDim.x + threadIdx.x;
  if (i >= E2) return;
  ea[i] = (i < E) ? ea_in[i] : ((*rsum) / (float)E);
}

__global__ void fill_f32(float* __restrict__ p, float v, int n)
{ int i = blockIdx.x * blockDim.x + threadIdx.x; if (i < n) p[i] = v; }

__global__ void fill_u32(unsigned* __restrict__ p, unsigned v, int n)
{ int i = blockIdx.x * blockDim.x + threadIdx.x; if (i < n) p[i] = v; }

// ---------------- GATv2 pieces ----------------
// logits: one wave per (edge, head); lanes cover the 64 channels (2 each)
__global__ void __launch_bounds__(256)
gat_logits(const float* __restrict__ xl, const float* __restrict__ xr,
           const float* __restrict__ ea, const int* __restrict__ src,
           const int* __restrict__ dst, const float* __restrict__ we,
           const float* __restrict__ att, float* __restrict__ logits,
           unsigned* __restrict__ maxbuf, int E2)
{
  const int gw   = blockIdx.x * (blockDim.x >> 5) + (threadIdx.x >> 5);
  const int lane = threadIdx.x & 31;
  const int e = gw >> 3, h = gw & 7;
  if (e >= E2) return;
  const int s = src[e], d = dst[e];
  const float eav = ea[e];
  const int base = h * 64;
  float acc = 0.f;
  #pragma unroll
  for (int j = 0; j < 2; ++j) {
    const int c = base + lane + j * 32;
    float v = xl[(size_t)s * HC + c] + xr[(size_t)d * HC + c] + eav * we[c];
    v = v > 0.f ? v : 0.2f * v;                 // leaky_relu(0.2)
    acc += v * att[c];
  }
  #pragma unroll
  for (int off = 16; off > 0; off >>= 1) acc += __shfl_xor(acc, off, 32);
  if (lane == 0) {
    logits[(size_t)e * 8 + h] = acc;
    atomicMax(&maxbuf[(size_t)d * 8 + h], fenc(acc));
  }
}

__global__ void gat_exp(float* __restrict__ logits, const int* __restrict__ dst,
                        const unsigned* __restrict__ maxbuf, float* __restrict__ sumbuf,
                        int E2)
{
  int i = blockIdx.x * blockDim.x + threadIdx.x;
  if (i >= E2 * 8) return;
  const int e = i >> 3, h = i & 7;
  const int d = dst[e];
  const float ex = expf(logits[i] - fdec(maxbuf[(size_t)d * 8 + h]));
  logits[i] = ex;
  atomicAdd(&sumbuf[(size_t)d * 8 + h], ex);
}

// scatter-add: one thread per (edge, channel)
__global__ void gat_agg(const float* __restrict__ xl, const float* __restrict__ expv,
                        const float* __restrict__ sumbuf, const int* __restrict__ src,
                        const int* __restrict__ dst, float* __restrict__ out, int E2)
{
  int i = blockIdx.x * blockDim.x + threadIdx.x;
  if (i >= E2 * HC) return;
  const int e = i >> 9, c = i & 511, h = c >> 6;
  const int s = src[e], d = dst[e];
  const float alpha = expv[(size_t)e * 8 + h] / sumbuf[(size_t)d * 8 + h];
  atomicAdd(&out[(size_t)d * HC + c], xl[(size_t)s * HC + c] * alpha);
}

__global__ void gat_finish(float* __restrict__ y, const float* __restrict__ bias, int n)
{
  int i = blockIdx.x * blockDim.x + threadIdx.x;
  if (i >= n) return;
  y[i] = gelu_f(y[i] + bias[i & 511]);
}

// ---------------- tiny final GEMM (N=4) + output assembly ----------------
__global__ void gemm_small(const float* __restrict__ X, const float* __restrict__ W,
                           const float* __restrict__ b, float* __restrict__ Y,
                           int Mrows, int N, int K)
{
  int i = blockIdx.x * blockDim.x + threadIdx.x;
  if (i >= Mrows * N) return;
  const int m = i / N, n = i - m * N;
  float acc = b[n];
  for (int k = 0; k < K; ++k) acc += X[(size_t)m * K + k] * W[(size_t)n * K + k];
  Y[i] = acc;
}

__global__ void assemble_out(const float* __restrict__ y4, const float* __restrict__ x,
                             float* __restrict__ out, int Mrows)
{
  int m = blockIdx.x * blockDim.x + threadIdx.x;
  if (m >= Mrows) return;
  out[m * 4 + 0] = y4[m * 4 + 0] + x[m * 2 + 0];   // mean x
  out[m * 4 + 1] = y4[m * 4 + 2];                  // log_std x
  out[m * 4 + 2] = y4[m * 4 + 1] + x[m * 2 + 1];   // mean y
  out[m * 4 + 3] = y4[m * 4 + 3];                  // log_std y
}

// ---------------- host orchestration ----------------
static inline unsigned nblk(size_t n, int t) { return (unsigned)((n + (size_t)t - 1) / t); }

extern "C" void kernel_launch(void* const* d_in, const int* in_sizes, int n_in,
                              void* d_out, int out_size, void* d_ws, size_t ws_size,
                              hipStream_t stream)
{
  (void)in_sizes; (void)n_in; (void)out_size; (void)ws_size;
  // ---- input index map (setup_inputs() dict insertion order, depth-first) ----
  // 0:x 1:edge_index 2:edge_attr 3:params 4:enc1_w 5:enc1_b 6:enc2_w 7:enc2_b
  // tenc layer l: base 8+12l -> wv,bv,wo,bo,w1,b1,w2,b2,n1g,n1b,n2g,n2b
  // 32:gr_w 33:gr_b ; gat layer l: base 34+7l -> wl,bl,wr,br,we,att,bias
  // 55:gr2_w 56:gr2_b ; tdec layer l: base 57+18l ->
  //   sa.wv,sa.bv,sa.wo,sa.bo,ca.wv,ca.bv,ca.wo,ca.bo,w1,b1,w2,b2,n1g,n1b,n2g,n2b,n3g,n3b
  // 93:dec1_w 94:dec1_b 95:dec2_w 96:dec2_b
  auto F = [&](int i) { return (const float*)d_in[i]; };
  const float* x_in   = F(0);
  const int*   e_idx  = (const int*)d_in[1];
  const float* e_attr = F(2);
  const float* params = F(3);

  // ---- workspace layout (float units, 256B aligned chunks) ----
  float* ws = (float*)d_ws;
  size_t off = 0;
  auto alloc = [&](size_t nf) { size_t o = off; off += (nf + 63) & ~(size_t)63; return ws + o; };
  float*    bufY   = alloc((size_t)M_ * HC);
  float*    bufXL  = alloc((size_t)M_ * HC);
  float*    bufXR  = alloc((size_t)M_ * HC);
  float*    encB   = alloc((size_t)M_ * HID);
  float*    s1     = alloc((size_t)M_ * HID);
  float*    s2     = alloc((size_t)M_ * HID);
  float*    s3     = alloc((size_t)M_ * HID);
  float*    h32    = alloc((size_t)M_ * FF);
  float*    xin    = alloc((size_t)M_ * 8);
  float*    y4     = alloc((size_t)M_ * 4);
  float*    logits = alloc((size_t)E2_ * 8);
  unsigned* maxb   = (unsigned*)alloc((size_t)M_ * 8);
  float*    sumb   = alloc((size_t)M_ * 8);
  float*    eaB    = alloc((size_t)E2_);
  int*      srcB   = (int*)alloc((size_t)E2_);
  int*      dstB   = (int*)alloc((size_t)E2_);
  float*    rsum   = alloc(1);

  auto gemm = [&](const float* X, int wI, int bI, float* Y, int Mr, int N, int K, int act) {
    const float* Wp = F(wI);
    const float* Bp = bI >= 0 ? F(bI) : nullptr;
    if ((N & 63) == 0 && (Mr & 63) == 0) {
      const int tiles = (Mr / 64) * (N / 64);   // one 64x64 tile per wave, 16 WMMA/K-step
      gemm_wmma_f32_blk<<<nblk(tiles, 8), 256, 0, stream>>>(X, Wp, Bp, Y, Mr, N, K, act);
    } else {
      const int tiles = (Mr / 16) * (N / 16);
      gemm_wmma_f32<<<nblk(tiles, 8), 256, 0, stream>>>(X, Wp, Bp, Y, Mr, N, K, act);
    }
  };
  auto add_ln = [&](float* dst, const float* b, int gI, int bI) {
    add_ln_kernel<<<nblk((size_t)M_ * 32, 256), 256, 0, stream>>>(dst, b, F(gI), F(bI), dst, M_);
  };

  // ---- encoder MLP ----
  build_xin<<<nblk((size_t)M_ * 8, 256), 256, 0, stream>>>(x_in, params, xin, M_);
  gemm(xin, 4, 5, s2, M_, HID, 8, ACT_GELU);     // enc1 + gelu
  gemm(s2, 6, 7, encB, M_, HID, HID, ACT_NONE);  // enc2
  // ---- 2 transformer-encoder layers (token-local) ----
  for (int l = 0; l < 2; ++l) {
    const int b = 8 + 12 * l;
    gemm(encB, b + 0, b + 1, s2, M_, HID, HID, ACT_NONE);   // wv
    gemm(s2,   b + 2, b + 3, s3, M_, HID, HID, ACT_NONE);   // wo
    add_ln(encB, s3, b + 8, b + 9);                         // n1
    gemm(encB, b + 4, b + 5, h32, M_, FF, HID, ACT_RELU);   // w1+relu (N=32 -> simple path)
    gemm(h32,  b + 6, b + 7, s3, M_, HID, FF, ACT_NONE);    // w2
    add_ln(encB, s3, b + 10, b + 11);                       // n2
  }
  // ---- project to graph width ----
  gemm(encB, 32, 33, bufY, M_, HC, HID, ACT_NONE);          // gr

  // ---- edge preprocessing (self loops + mean edge attr) ----
  build_edges<<<nblk(E2_, 256), 256, 0, stream>>>(e_idx, srcB, dstB, E_, E2_);
  fill_f32<<<1, 32, 0, stream>>>(rsum, 0.f, 1);
  reduce_sum<<<nblk(E_, 256), 256, 0, stream>>>(e_attr, rsum, E_);
  build_ea<<<nblk(E2_, 256), 256, 0, stream>>>(e_attr, rsum, eaB, E_, E2_);

  // ---- 3 GATv2 layers ----
  for (int l = 0; l < 3; ++l) {
    const int b = 34 + 7 * l;
    gemm(bufY, b + 0, b + 1, bufXL, M_, HC, HC, ACT_NONE);  // wl
    gemm(bufY, b + 2, b + 3, bufXR, M_, HC, HC, ACT_NONE);  // wr
    fill_u32<<<nblk((size_t)M_ * 8, 256), 256, 0, stream>>>(maxb, 0x007fffffu /*enc(-inf)*/, M_ * 8);
    fill_f32<<<nblk((size_t)M_ * 8, 256), 256, 0, stream>>>(sumb, 0.f, M_ * 8);
    gat_logits<<<nblk((size_t)E2_ * 8 * 32, 256), 256, 0, stream>>>(
        bufXL, bufXR, eaB, srcB, dstB, F(b + 4), F(b + 5), logits, maxb, E2_);
    gat_exp<<<nblk((size_t)E2_ * 8, 256), 256, 0, stream>>>(logits, dstB, maxb, sumb, E2_);
    fill_f32<<<nblk((size_t)M_ * HC, 256), 256, 0, stream>>>(bufY, 0.f, M_ * HC);
    gat_agg<<<nblk((size_t)E2_ * HC, 256), 256, 0, stream>>>(bufXL, logits, sumb, srcB, dstB, bufY, E2_);
    gat_finish<<<nblk((size_t)M_ * HC, 256), 256, 0, stream>>>(bufY, F(b + 6), M_ * HC);
  }

  // ---- back to 64 ch + 2 decoder layers ----
  gemm(bufY, 55, 56, s1, M_, HID, HC, ACT_NONE);            // gr2
  for (int l = 0; l < 2; ++l) {
    const int b = 57 + 18 * l;
    gemm(s1,   b + 0, b + 1, s2, M_, HID, HID, ACT_NONE);   // sa.wv
    gemm(s2,   b + 2, b + 3, s3, M_, HID, HID, ACT_NONE);   // sa.wo
    add_ln(s1, s3, b + 12, b + 13);                         // n1
    gemm(encB, b + 4, b + 5, s2, M_, HID, HID, ACT_NONE);   // ca.wv (on memory = enc)
    gemm(s2,   b + 6, b + 7, s3, M_, HID, HID, ACT_NONE);   // ca.wo
    add_ln(s1, s3, b + 14, b + 15);                         // n2
    gemm(s1,   b + 8, b + 9, h32, M_, FF, HID, ACT_RELU);   // w1+relu
    gemm(h32,  b + 10, b + 11, s3, M_, HID, FF, ACT_NONE);  // w2
    add_ln(s1, s3, b + 16, b + 17);                         // n3
  }

  // ---- head ----
  gemm(s1, 93, 94, s2, M_, HID, HID, ACT_GELU);             // dec1 + gelu
  gemm_small<<<nblk((size_t)M_ * 4, 256), 256, 0, stream>>>(s2, F(95), F(96), y4, M_, 4, HID);
  assemble_out<<<nblk(M_, 256), 256, 0, stream>>>(y4, x_in, (float*)d_out, M_);
}